// GATEdgeModel_11793980195349
// MI455X (gfx1250) — compile-verified
//
#include <hip/hip_runtime.h>
#include <cstddef>

#define N_NODES 10000
#define N_EDGES 160000
#define F_IN    128
#define E_IN    32
#define FDIM    64
#define HEADS   4
#define OUTC    64
#define GRAPHS  64

typedef float v2f __attribute__((ext_vector_type(2)));
typedef float v8f __attribute__((ext_vector_type(8)));

__device__ __forceinline__ float elu_f(float x) { return x > 0.f ? x : (expf(x) - 1.f); }

// order-preserving bijection float -> uint for atomicMax-based float max
__device__ __forceinline__ unsigned f2ord(float x) {
  unsigned u = __float_as_uint(x);
  return (u & 0x80000000u) ? ~u : (u | 0x80000000u);
}
__device__ __forceinline__ float ord2f(unsigned u) {
  return (u & 0x80000000u) ? __uint_as_float(u & 0x7FFFFFFFu) : __uint_as_float(~u);
}

__global__ void fill_u32_kernel(unsigned* __restrict__ p, unsigned v, int n) {
  int i = blockIdx.x * blockDim.x + threadIdx.x;
  if (i < n) p[i] = v;
}

// ---------------------------------------------------------------------------
// C[M,Ncols] = A[M,K] @ W[K,Ncols], f32, one wave per 16x16 output tile.
// WMMA f32 16x16x4 fragment layouts (wave32):
//   A: lane l, vgpr v -> (row = 16*tM + (l&15), k = kk + v + 2*(l>>4))
//   B: lane l, vgpr v -> (k = kk + v + 2*(l>>4), col = 16*tN + (l&15))
//   C: lane l, vgpr i -> (row = 16*tM + i + 8*(l>>4), col = 16*tN + (l&15))
// M, Ncols multiples of 16 and K multiple of 4 for every GEMM in this model.
// ---------------------------------------------------------------------------
__global__ void wmma_gemm_f32_kernel(const float* __restrict__ A, const float* __restrict__ W,
                                     float* __restrict__ C, int M, int K, int Ncols) {
  int wavesPerBlock = blockDim.x >> 5;
  int wave = blockIdx.x * wavesPerBlock + (threadIdx.x >> 5);
  int lane = threadIdx.x & 31;
  int tilesN = Ncols >> 4;
  int tileM = wave / tilesN;
  int tileN = wave - tileM * tilesN;
  if (tileM * 16 >= M) return;              // wave-uniform exit: EXEC stays all-ones
  int half = lane >> 4;                     // 0 or 1
  int l16  = lane & 15;
  const float* arow = A + (size_t)(tileM * 16 + l16) * K + half * 2;
  const float* wcol = W + (size_t)(half * 2) * Ncols + tileN * 16 + l16;
  v8f acc = {};
  for (int kk = 0; kk < K; kk += 4) {
    v2f a, b;
    a.x = arow[kk];
    a.y = arow[kk + 1];
    b.x = wcol[(size_t)kk * Ncols];
    b.y = wcol[(size_t)(kk + 1) * Ncols];
    acc = __builtin_amdgcn_wmma_f32_16x16x4_f32(false, a, false, b, (short)0, acc, false, false);
  }
  float* crow = C + (size_t)(tileM * 16 + half * 8) * Ncols + tileN * 16 + l16;
#pragma unroll
  for (int i = 0; i < 8; ++i) crow[(size_t)i * Ncols] = acc[i];
}

// ---------------------------------------------------------------------------
// Per-edge attention logits + running segment-max over dst.
// One wave per edge. lanes split into H groups of 32/H lanes; each lane
// handles F/(32/H) contiguous features of its head.
// ---------------------------------------------------------------------------
template <int H, int F>
__global__ void edge_logits_kernel(const float* __restrict__ h, const float* __restrict__ ep,
                                   const int* __restrict__ src, const int* __restrict__ dst,
                                   const float* __restrict__ a, float* __restrict__ lg,
                                   unsigned* __restrict__ nmax, int E) {
  constexpr int HF  = H * F;
  constexpr int LPH = 32 / H;   // lanes per head
  constexpr int EPL = F / LPH;  // elems per lane
  int e = blockIdx.x * (blockDim.x >> 5) + (threadIdx.x >> 5);
  if (e >= E) return;
  int lane = threadIdx.x & 31;
  int head = lane / LPH;
  int f0   = (lane % LPH) * EPL;
  int s = src[e], d = dst[e];
  const float* hs = h  + (size_t)s * HF + head * F + f0;
  const float* hd = h  + (size_t)d * HF + head * F + f0;
  const float* pe = ep + (size_t)e * HF + head * F + f0;
  const float* a0 = a + head * F + f0;
  const float* a1 = a + HF + head * F + f0;
  const float* a2 = a + 2 * HF + head * F + f0;
  float acc = 0.f;
#pragma unroll
  for (int j = 0; j < EPL; ++j)
    acc += hs[j] * a0[j] + hd[j] * a1[j] + pe[j] * a2[j];
#pragma unroll
  for (int off = 1; off < LPH; off <<= 1) acc += __shfl_xor(acc, off, 32);
  if ((lane % LPH) == 0) {
    float l = (acc > 0.f ? acc : 0.2f * acc);   // leaky_relu, SCALING = 1.0
    lg[(size_t)e * H + head] = l;
    atomicMax(&nmax[(size_t)d * H + head], f2ord(l));
  }
}

// exp(logit - segmax) and segment-sum of denominators. One thread per (edge,head).
__global__ void edge_exp_kernel(float* __restrict__ lg, const int* __restrict__ dst,
                                const unsigned* __restrict__ nmax, float* __restrict__ nden,
                                int E, int H) {
  int i = blockIdx.x * blockDim.x + threadIdx.x;
  if (i >= E * H) return;
  int e = i / H, hh = i - e * H;
  int d = dst[e];
  float ex = expf(lg[i] - ord2f(nmax[(size_t)d * H + hh]));
  lg[i] = ex;
  atomicAdd(&nden[(size_t)d * H + hh], ex);
}

// alpha-weighted message scatter-add + edge_out = mean_h(hs+hd+ep) (+ELU).
template <int H, int F, bool ACT>
__global__ void edge_msg_kernel(const float* __restrict__ h, const float* __restrict__ ep,
                                const float* __restrict__ lg, const float* __restrict__ nden,
                                const int* __restrict__ src, const int* __restrict__ dst,
                                float* __restrict__ nacc, float* __restrict__ e_out, int E) {
  constexpr int HF  = H * F;
  constexpr int LPH = 32 / H;
  constexpr int EPL = F / LPH;
  int e = blockIdx.x * (blockDim.x >> 5) + (threadIdx.x >> 5);
  if (e >= E) return;
  int lane = threadIdx.x & 31;
  int head = lane / LPH;
  int f0   = (lane % LPH) * EPL;
  int s = src[e], d = dst[e];
  const float* hs = h  + (size_t)s * HF + head * F + f0;
  const float* hd = h  + (size_t)d * HF + head * F + f0;
  const float* pe = ep + (size_t)e * HF + head * F + f0;
  float alpha = lg[(size_t)e * H + head] / nden[(size_t)d * H + head];
  float eo[EPL];
#pragma unroll
  for (int j = 0; j < EPL; ++j) {
    float vs = hs[j], vd = hd[j], ve = pe[j];
    atomicAdd(&nacc[(size_t)d * HF + head * F + f0 + j], alpha * (vs + ve));
    eo[j] = vs + vd + ve;
  }
  // sum over heads: xor over the head-selecting lane bits
#pragma unroll
  for (int off = LPH; off < 32; off <<= 1)
#pragma unroll
    for (int j = 0; j < EPL; ++j) eo[j] += __shfl_xor(eo[j], off, 32);
  if (head == 0) {
    float* er = e_out + (size_t)e * F + f0;
#pragma unroll
    for (int j = 0; j < EPL; ++j) {
      float v = eo[j] * (1.0f / H);
      if (ACT) v = elu_f(v);
      er[j] = v;
    }
  }
}

template <int H, int F, bool ACT>
__global__ void node_finalize_kernel(const float* __restrict__ nacc, float* __restrict__ xo, int N) {
  int i = blockIdx.x * blockDim.x + threadIdx.x;
  if (i >= N * F) return;
  int n = i / F, f = i - n * F;
  float sum = 0.f;
#pragma unroll
  for (int hh = 0; hh < H; ++hh) sum += nacc[(size_t)n * H * F + hh * F + f];
  float v = sum * (1.0f / H);
  if (ACT) v = elu_f(v);
  xo[i] = v;
}

__global__ void readout_sum_kernel(const float* __restrict__ x, const int* __restrict__ batch,
                                   float* __restrict__ sums, float* __restrict__ cnts, int N, int C) {
  int i = blockIdx.x * blockDim.x + threadIdx.x;
  if (i >= N * C) return;
  int n = i / C, f = i - n * C;
  int g = batch[n];
  atomicAdd(&sums[(size_t)g * C + f], x[i]);
  if (f == 0) atomicAdd(&cnts[g], 1.0f);
}

__global__ void readout_div_kernel(const float* __restrict__ sums, const float* __restrict__ cnts,
                                   float* __restrict__ out, int GC, int C) {
  int i = blockIdx.x * blockDim.x + threadIdx.x;
  if (i >= GC) return;
  out[i] = sums[i] / fmaxf(cnts[i / C], 1.0f);
}

// ---------------------------------------------------------------------------
// Host-side per-layer driver (all launches on `stream`, capture-safe).
// ---------------------------------------------------------------------------
template <int H, int F, bool ACT>
static void run_layer(const float* x_in, int Fin, const float* e_in, int Ein,
                      const float* Wn, const float* We, const float* a,
                      float* h, float* ep, float* lg, unsigned* nmax, float* nden, float* nacc,
                      float* x_out, float* e_out, const int* src, const int* dst,
                      hipStream_t stream) {
  const int HF = H * F;
  {
    int n1 = N_NODES * H;
    fill_u32_kernel<<<(n1 + 255) / 256, 256, 0, stream>>>(nmax, 0u, n1);
    fill_u32_kernel<<<(n1 + 255) / 256, 256, 0, stream>>>((unsigned*)nden, 0u, n1);
    int n2 = N_NODES * HF;
    fill_u32_kernel<<<(n2 + 255) / 256, 256, 0, stream>>>((unsigned*)nacc, 0u, n2);
  }
  {
    int tilesN = (N_NODES / 16) * (HF / 16);
    wmma_gemm_f32_kernel<<<(tilesN + 3) / 4, 128, 0, stream>>>(x_in, Wn, h, N_NODES, Fin, HF);
    int tilesE = (N_EDGES / 16) * (HF / 16);
    wmma_gemm_f32_kernel<<<(tilesE + 3) / 4, 128, 0, stream>>>(e_in, We, ep, N_EDGES, Ein, HF);
  }
  edge_logits_kernel<H, F><<<(N_EDGES + 7) / 8, 256, 0, stream>>>(h, ep, src, dst, a, lg, nmax, N_EDGES);
  {
    int n = N_EDGES * H;
    edge_exp_kernel<<<(n + 255) / 256, 256, 0, stream>>>(lg, dst, nmax, nden, N_EDGES, H);
  }
  edge_msg_kernel<H, F, ACT><<<(N_EDGES + 7) / 8, 256, 0, stream>>>(h, ep, lg, nden, src, dst, nacc, e_out, N_EDGES);
  {
    int n = N_NODES * F;
    node_finalize_kernel<H, F, ACT><<<(n + 255) / 256, 256, 0, stream>>>(nacc, x_out, N_NODES);
  }
}

extern "C" void kernel_launch(void* const* d_in, const int* in_sizes, int n_in,
                              void* d_out, int out_size, void* d_ws, size_t ws_size,
                              hipStream_t stream) {
  (void)in_sizes; (void)n_in; (void)out_size; (void)ws_size;
  const float* x0     = (const float*)d_in[0];
  const float* e0     = (const float*)d_in[1];
  const float* Wn0    = (const float*)d_in[2];
  const float* We0    = (const float*)d_in[3];
  const float* a0     = (const float*)d_in[4];
  const float* Wn_mid = (const float*)d_in[5];
  const float* We_mid = (const float*)d_in[6];
  const float* a_mid  = (const float*)d_in[7];
  const float* Wn_out = (const float*)d_in[8];
  const float* We_out = (const float*)d_in[9];
  const float* a_out  = (const float*)d_in[10];
  const int*   eidx   = (const int*)d_in[11];
  const int*   batch  = (const int*)d_in[12];
  const int* src = eidx;
  const int* dst = eidx + N_EDGES;

  // workspace layout (~231 MB of f32)
  float* ws = (float*)d_ws;
  size_t off = 0;
  float*    h    = ws + off; off += (size_t)N_NODES * HEADS * FDIM;   // [N,256]
  float*    ep   = ws + off; off += (size_t)N_EDGES * HEADS * FDIM;   // [E,256]
  float*    lg   = ws + off; off += (size_t)N_EDGES * HEADS;          // [E,4]
  unsigned* nmax = (unsigned*)(ws + off); off += (size_t)N_NODES * HEADS;
  float*    nden = ws + off; off += (size_t)N_NODES * HEADS;
  float*    nacc = ws + off; off += (size_t)N_NODES * HEADS * FDIM;   // [N,256]
  float*    xbuf = ws + off; off += (size_t)N_NODES * FDIM;           // [N,64]
  float*    ebuf = ws + off; off += (size_t)N_EDGES * FDIM;           // [E,64]
  float*    sums = ws + off; off += (size_t)GRAPHS * OUTC;
  float*    cnts = ws + off; off += GRAPHS;

  // layer 0
  run_layer<HEADS, FDIM, true>(x0, F_IN, e0, E_IN, Wn0, We0, a0,
                               h, ep, lg, nmax, nden, nacc, xbuf, ebuf, src, dst, stream);
  // middle layers (Wn_mid: [L,64,4,64] stride 64*256; a_mid: [L,3,4,64] stride 768)
  run_layer<HEADS, FDIM, true>(xbuf, FDIM, ebuf, FDIM, Wn_mid, We_mid, a_mid,
                               h, ep, lg, nmax, nden, nacc, xbuf, ebuf, src, dst, stream);
  run_layer<HEADS, FDIM, true>(xbuf, FDIM, ebuf, FDIM,
                               Wn_mid + 64 * 256, We_mid + 64 * 256, a_mid + 3 * 256,
                               h, ep, lg, nmax, nden, nacc, xbuf, ebuf, src, dst, stream);
  // output layer: 1 head, no activation
  run_layer<1, OUTC, false>(xbuf, FDIM, ebuf, FDIM, Wn_out, We_out, a_out,
                            h, ep, lg, nmax, nden, nacc, xbuf, ebuf, src, dst, stream);

  // graph readout: scatter-mean over batch_vector
  {
    int nz = GRAPHS * OUTC + GRAPHS;
    fill_u32_kernel<<<(nz + 255) / 256, 256, 0, stream>>>((unsigned*)sums, 0u, nz);
    int n = N_NODES * OUTC;
    readout_sum_kernel<<<(n + 255) / 256, 256, 0, stream>>>(xbuf, batch, sums, cnts, N_NODES, OUTC);
    int gc = GRAPHS * OUTC;
    readout_div_kernel<<<(gc + 255) / 256, 256, 0, stream>>>(sums, cnts, (float*)d_out, gc, OUTC);
  }
}